// BongardGNN_22385369547064
// MI455X (gfx1250) — compile-verified
//
#include <hip/hip_runtime.h>
#include <hip/hip_bf16.h>

typedef __attribute__((ext_vector_type(2))) float v2f;
typedef __attribute__((ext_vector_type(8))) float v8f;

// ---------------------------------------------------------------------------
// Degree / normalization
// ---------------------------------------------------------------------------
__global__ void gcn_init_deg(unsigned int* __restrict__ deg, int n) {
    int i = blockIdx.x * blockDim.x + threadIdx.x;
    if (i < n) deg[i] = 1u;  // self-loop
}

__global__ void gcn_count_deg(unsigned int* __restrict__ deg,
                              const long long* __restrict__ ei, int E) {
    int e = blockIdx.x * blockDim.x + threadIdx.x;
    if (e >= E) return;
    int dst = (int)ei[E + e];
    atomicAdd(&deg[dst], 1u);
}

__global__ void gcn_finalize_dis(float* __restrict__ dis, int n) {
    int i = blockIdx.x * blockDim.x + threadIdx.x;
    if (i >= n) return;
    float d = (float)((const unsigned int*)dis)[i];
    float r = rsqrtf(d);
    r = r * (1.5f - 0.5f * d * r * r);  // one Newton step for fp32 accuracy
    dis[i] = r;
}

// ---------------------------------------------------------------------------
// GEMM1: h1[N,32] = x[N,16] @ W1[16,32]  (fp32 WMMA 16x16x4)
// One wave per 16x16 output tile; wave = rowTile*2 + colTile.
// A layout (16x4 f32): lanes 0-15 M=0..15; VGPR0 = K even-pair lo, VGPR1 = +1;
// lane half kh selects K offset {0,1} vs {2,3} within each 4-wide K step.
// B/C/D: row striped across lanes within a VGPR (C/D: VGPR r = rows r, r+8).
// ---------------------------------------------------------------------------
__global__ void gcn_gemm1_wmma(const float* __restrict__ x,
                               const float* __restrict__ W1,
                               float* __restrict__ h1, int nRowTiles) {
    int wave = (int)((blockIdx.x * blockDim.x + threadIdx.x) >> 5);
    int lane = threadIdx.x & 31;
    if (wave >= nRowTiles * 2) return;   // whole-wave uniform exit (EXEC all-1 in WMMA waves)
    int rowTile = wave >> 1;
    int nt      = wave & 1;              // column tile: 0 -> cols 0..15, 1 -> 16..31
    int m  = lane & 15;
    int kh = lane >> 4;
    int row = rowTile * 16 + m;
    int col = nt * 16 + m;

    v8f c = {};
#pragma unroll
    for (int kk = 0; kk < 16; kk += 4) {
        int k0 = kk + 2 * kh;
        v2f a;
        a.x = x[row * 16 + k0];
        a.y = x[row * 16 + k0 + 1];
        v2f b;
        b.x = W1[k0 * 32 + col];
        b.y = W1[(k0 + 1) * 32 + col];
        c = __builtin_amdgcn_wmma_f32_16x16x4_f32(false, a, false, b,
                                                  (short)0, c, false, false);
    }
    int outBase = (rowTile * 16 + 8 * kh) * 32 + col;
#pragma unroll
    for (int r = 0; r < 8; ++r)
        h1[outBase + r * 32] = c[r];
}

// Scalar tail for rows not covered by full 16-row tiles (not launched when N%16==0)
__global__ void gcn_gemm1_tail(const float* __restrict__ x, const float* __restrict__ W1,
                               float* __restrict__ h1, int rowStart, int nRows) {
    int t = blockIdx.x * blockDim.x + threadIdx.x;
    if (t >= nRows * 32) return;
    int row = rowStart + t / 32, col = t % 32;
    float s = 0.f;
    for (int k = 0; k < 16; ++k) s += x[row * 16 + k] * W1[k * 32 + col];
    h1[row * 32 + col] = s;
}

// ---------------------------------------------------------------------------
// Layer-1 aggregation
// ---------------------------------------------------------------------------
__global__ void gcn_init_acc1(float* __restrict__ acc1, const float* __restrict__ h1,
                              const float* __restrict__ dis, const float* __restrict__ b1,
                              int n) {
    int t = blockIdx.x * blockDim.x + threadIdx.x;
    if (t >= n * 32) return;
    int i = t >> 5, f = t & 31;
    float d = dis[i];
    acc1[t] = h1[t] * d * d + b1[f];
}

// 8 threads per edge, 4 features each (float4 gather from L2-resident h1)
__global__ void gcn_scatter1(float* __restrict__ acc1, const float* __restrict__ h1,
                             const float* __restrict__ dis,
                             const long long* __restrict__ ei, int E) {
    int t = blockIdx.x * blockDim.x + threadIdx.x;
    if (t >= E * 8) return;
    int e = t >> 3, f0 = (t & 7) * 4;
    int src = (int)ei[e];
    int dst = (int)ei[E + e];
    float w = dis[src] * dis[dst];
    const float4 hv = *reinterpret_cast<const float4*>(h1 + src * 32 + f0);
    float* p = acc1 + dst * 32 + f0;
    atomicAdd(p + 0, hv.x * w);
    atomicAdd(p + 1, hv.y * w);
    atomicAdd(p + 2, hv.z * w);
    atomicAdd(p + 3, hv.w * w);
}

// ---------------------------------------------------------------------------
// GEMM2: h2[N,2] = relu(acc1)[N,32] @ W2[32,2], W2 zero-padded to 32x16
// ---------------------------------------------------------------------------
__global__ void gcn_pad_w2(float* __restrict__ w2p, const float* __restrict__ W2) {
    int t = blockIdx.x * blockDim.x + threadIdx.x;
    if (t >= 32 * 16) return;
    int k = t >> 4, ccol = t & 15;
    w2p[t] = (ccol < 2) ? W2[k * 2 + ccol] : 0.0f;
}

__global__ void gcn_gemm2_wmma(const float* __restrict__ acc1,
                               const float* __restrict__ w2p,
                               float* __restrict__ h2, int nRowTiles) {
    int wave = (int)((blockIdx.x * blockDim.x + threadIdx.x) >> 5);
    int lane = threadIdx.x & 31;
    if (wave >= nRowTiles) return;
    int m  = lane & 15;
    int kh = lane >> 4;
    int row = wave * 16 + m;

    v8f c = {};
#pragma unroll
    for (int kk = 0; kk < 32; kk += 4) {
        int k0 = kk + 2 * kh;
        float a0 = acc1[row * 32 + k0];
        float a1 = acc1[row * 32 + k0 + 1];
        a0 = a0 > 0.f ? a0 : 0.f;   // fused ReLU
        a1 = a1 > 0.f ? a1 : 0.f;
        v2f a; a.x = a0; a.y = a1;
        v2f b;
        b.x = w2p[k0 * 16 + m];
        b.y = w2p[(k0 + 1) * 16 + m];
        c = __builtin_amdgcn_wmma_f32_16x16x4_f32(false, a, false, b,
                                                  (short)0, c, false, false);
    }
    if (m < 2) {                    // only N=0,1 are real columns
        int base = (wave * 16 + 8 * kh) * 2 + m;
#pragma unroll
        for (int r = 0; r < 8; ++r) h2[base + r * 2] = c[r];
    }
}

__global__ void gcn_gemm2_tail(const float* __restrict__ acc1, const float* __restrict__ W2,
                               float* __restrict__ h2, int rowStart, int nRows) {
    int t = blockIdx.x * blockDim.x + threadIdx.x;
    if (t >= nRows * 2) return;
    int row = rowStart + (t >> 1), col = t & 1;
    float s = 0.f;
    for (int k = 0; k < 32; ++k) {
        float a = acc1[row * 32 + k];
        a = a > 0.f ? a : 0.f;
        s += a * W2[k * 2 + col];
    }
    h2[row * 2 + col] = s;
}

// ---------------------------------------------------------------------------
// Layer-2 aggregation -> d_out
// ---------------------------------------------------------------------------
__global__ void gcn_init_out(float* __restrict__ out, const float* __restrict__ h2,
                             const float* __restrict__ dis, const float* __restrict__ b2,
                             int n) {
    int t = blockIdx.x * blockDim.x + threadIdx.x;
    if (t >= n * 2) return;
    int i = t >> 1, f = t & 1;
    float d = dis[i];
    out[t] = h2[t] * d * d + b2[f];
}

__global__ void gcn_scatter2(float* __restrict__ out, const float* __restrict__ h2,
                             const float* __restrict__ dis,
                             const long long* __restrict__ ei, int E) {
    int e = blockIdx.x * blockDim.x + threadIdx.x;
    if (e >= E) return;
    int src = (int)ei[e];
    int dst = (int)ei[E + e];
    float w = dis[src] * dis[dst];
    atomicAdd(&out[dst * 2 + 0], h2[src * 2 + 0] * w);
    atomicAdd(&out[dst * 2 + 1], h2[src * 2 + 1] * w);
}

// ---------------------------------------------------------------------------
// Launch
// ---------------------------------------------------------------------------
extern "C" void kernel_launch(void* const* d_in, const int* in_sizes, int n_in,
                              void* d_out, int out_size, void* d_ws, size_t ws_size,
                              hipStream_t stream) {
    const float*     x  = (const float*)d_in[0];
    const long long* ei = (const long long*)d_in[1];   // int64 [2, E]
    const float*     W1 = (const float*)d_in[2];
    const float*     b1 = (const float*)d_in[3];
    const float*     W2 = (const float*)d_in[4];
    const float*     b2 = (const float*)d_in[5];
    float* out = (float*)d_out;

    const int N = in_sizes[0] / 16;       // 200000
    const int E = in_sizes[1] / 2;        // 3200000

    // workspace layout (256B aligned slices)
    char* ws = (char*)d_ws;
    size_t off = 0;
    auto take = [&](size_t bytes) -> char* {
        char* p = ws + off;
        off += (bytes + 255) & ~(size_t)255;
        return p;
    };
    float* dis  = (float*)take((size_t)N * 4);        // degree (uint) then rsqrt
    float* h1   = (float*)take((size_t)N * 32 * 4);
    float* acc1 = (float*)take((size_t)N * 32 * 4);
    float* h2   = (float*)take((size_t)N * 2 * 4);
    float* w2p  = (float*)take(32 * 16 * 4);

    const int BS = 256;
    auto blocks = [&](long long n) { return (int)((n + BS - 1) / BS); };

    // 1. degrees -> dis
    gcn_init_deg<<<blocks(N), BS, 0, stream>>>((unsigned int*)dis, N);
    gcn_count_deg<<<blocks(E), BS, 0, stream>>>((unsigned int*)dis, ei, E);
    gcn_finalize_dis<<<blocks(N), BS, 0, stream>>>(dis, N);

    // 2. h1 = x @ W1 (WMMA)
    int nRowTiles = N / 16;
    int rem = N - nRowTiles * 16;
    gcn_gemm1_wmma<<<blocks((long long)nRowTiles * 2 * 32), BS, 0, stream>>>(x, W1, h1, nRowTiles);
    if (rem > 0)
        gcn_gemm1_tail<<<blocks((long long)rem * 32), BS, 0, stream>>>(x, W1, h1, nRowTiles * 16, rem);

    // 3. layer-1 aggregation
    gcn_init_acc1<<<blocks((long long)N * 32), BS, 0, stream>>>(acc1, h1, dis, b1, N);
    gcn_scatter1<<<blocks((long long)E * 8), BS, 0, stream>>>(acc1, h1, dis, ei, E);

    // 4. h2 = relu(acc1) @ W2 (WMMA, W2 padded to 32x16)
    gcn_pad_w2<<<2, BS, 0, stream>>>(w2p, W2);
    gcn_gemm2_wmma<<<blocks((long long)nRowTiles * 32), BS, 0, stream>>>(acc1, w2p, h2, nRowTiles);
    if (rem > 0)
        gcn_gemm2_tail<<<blocks((long long)rem * 2), BS, 0, stream>>>(acc1, W2, h2, nRowTiles * 16, rem);

    // 5. layer-2 aggregation into d_out
    gcn_init_out<<<blocks((long long)N * 2), BS, 0, stream>>>(out, h2, dis, b2, N);
    gcn_scatter2<<<blocks(E), BS, 0, stream>>>(out, h2, dis, ei, E);
}